// HeteroGNN_43645457662174
// MI455X (gfx1250) — compile-verified
//
#include <hip/hip_runtime.h>

typedef __attribute__((ext_vector_type(2))) float v2f;
typedef __attribute__((ext_vector_type(8))) float v8f;

// ---------------------------------------------------------------------------
// Guaranteed single-instruction fp32 atomic add (no return -> STOREcnt).
// Lowers to GLOBAL_ATOMIC_ADD_F32 (ISA §15.18.3 opcode 86), resolved in L2.
// ---------------------------------------------------------------------------
__device__ __forceinline__ void atomic_add_f32(float* p, float v)
{
    asm volatile("global_atomic_add_f32 %0, %1, off"
                 :: "v"(p), "v"(v) : "memory");
}

// ---------------------------------------------------------------------------
// Fused GEMM  xw[N,64] = x[N,64] @ [W_near | W_similar]  (f32 WMMA)
// One wave per 16x16 output tile; 4 waves/block cover the 4 column tiles.
// ---------------------------------------------------------------------------
__global__ __launch_bounds__(128)
void gemm_xw_kernel(const float* __restrict__ x,
                    const float* __restrict__ Wn,
                    const float* __restrict__ Ws,
                    float* __restrict__ xw, int nrows)
{
    const int lane  = threadIdx.x & 31;
    const int wave  = threadIdx.x >> 5;        // 0..3 -> output column tile
    const int m0    = blockIdx.x * 16;         // row tile base
    const int lhalf = lane >> 4;               // 0: lanes 0-15, 1: lanes 16-31
    const int lm    = lane & 15;

    const float* __restrict__ W = (wave < 2) ? Wn : Ws;
    const int n0 = (wave & 1) * 16;            // column offset inside W (width 32)

    int arow = m0 + lm;                        // A-matrix row for this lane
    if (arow >= nrows) arow = nrows - 1;       // clamp: keep EXEC all-ones

    v8f c = {};
    #pragma unroll
    for (int k0 = 0; k0 < 64; k0 += 4) {
        // A 16x4 f32 layout: VGPR0 = K = 2*lhalf, VGPR1 = K = 2*lhalf+1
        v2f a = *(const v2f*)(x + (size_t)arow * 64 + k0 + 2 * lhalf);
        // B 4x16 f32 layout: row K striped across lanes within a VGPR
        v2f b;
        b.x = W[(k0 + 2 * lhalf)     * 32 + n0 + lm];
        b.y = W[(k0 + 2 * lhalf + 1) * 32 + n0 + lm];
        c = __builtin_amdgcn_wmma_f32_16x16x4_f32(false, a, false, b,
                                                  (short)0, c, false, false);
    }

    // C/D 16x16 f32: VGPR v -> M = v + 8*lhalf, N = lane%16
    const int gcol = wave * 16 + lm;           // global column in xw (0..63)
    #pragma unroll
    for (int v = 0; v < 8; ++v) {
        const int row = m0 + v + 8 * lhalf;
        if (row < nrows) xw[(size_t)row * 64 + gcol] = c[v];
    }
}

// ---------------------------------------------------------------------------
// Degrees start at 1.0 (self-loop included)
// ---------------------------------------------------------------------------
__global__ void init_deg_kernel(float* __restrict__ dn, float* __restrict__ ds, int n)
{
    const int i = blockIdx.x * blockDim.x + threadIdx.x;
    if (i < n) { dn[i] = 1.0f; ds[i] = 1.0f; }
}

// ---------------------------------------------------------------------------
// In-degree accumulation for both edge types (blockIdx.y selects type)
// ---------------------------------------------------------------------------
__global__ void count_deg_kernel(const int* __restrict__ dst_n,
                                 const int* __restrict__ dst_s,
                                 float* __restrict__ dn,
                                 float* __restrict__ ds, int E)
{
    const int e = blockIdx.x * blockDim.x + threadIdx.x;
    if (e >= E) return;
    const int*   dst = blockIdx.y ? dst_s : dst_n;
    float*       deg = blockIdx.y ? ds    : dn;
    atomic_add_f32(&deg[dst[e]], 1.0f);
}

// ---------------------------------------------------------------------------
// deg -> d^{-1/2} in place (deg >= 1 always, guard kept for parity)
// ---------------------------------------------------------------------------
__global__ void rsqrt_kernel(float* __restrict__ dn, float* __restrict__ ds, int n)
{
    const int i = blockIdx.x * blockDim.x + threadIdx.x;
    if (i < n) {
        const float a = dn[i], b = ds[i];
        dn[i] = (a > 0.0f) ? rsqrtf(a) : 0.0f;
        ds[i] = (b > 0.0f) ? rsqrtf(b) : 0.0f;
    }
}

// ---------------------------------------------------------------------------
// h[i,j] = self-loop messages for both edge types (norm = dinv[i]^2)
// ---------------------------------------------------------------------------
__global__ void init_h_kernel(const float* __restrict__ xw,
                              const float* __restrict__ dn,
                              const float* __restrict__ ds,
                              float* __restrict__ h, int n)
{
    const int t = blockIdx.x * blockDim.x + threadIdx.x;
    if (t >= n * 32) return;
    const int i = t >> 5, j = t & 31;
    const float a = dn[i], b = ds[i];
    h[t] = xw[(size_t)i * 64 + j]      * (a * a)
         + xw[(size_t)i * 64 + 32 + j] * (b * b);
}

// ---------------------------------------------------------------------------
// Edge scatter for both edge types (blockIdx.y selects type):
//   h[dst,j] += xw[src, cbase+j] * dinv[src]*dinv[dst]
// One wave per edge, 32 lanes = 32 features (128B coalesced gather + atomic).
// ---------------------------------------------------------------------------
__global__ void scatter_kernel(const int* __restrict__ en,
                               const int* __restrict__ es,
                               const float* __restrict__ xw,
                               const float* __restrict__ dn,
                               const float* __restrict__ dsv,
                               float* __restrict__ h, int E)
{
    const unsigned t = blockIdx.x * blockDim.x + threadIdx.x;
    const unsigned e = t >> 5;
    const int      j = (int)(t & 31u);
    if (e >= (unsigned)E) return;

    const int    typ   = blockIdx.y;
    const int*   edges = typ ? es  : en;
    const float* dinv  = typ ? dsv : dn;
    const int    cbase = typ ? 32  : 0;

    const int s = edges[e], d = edges[E + e];
    const float norm = dinv[s] * dinv[d];
    const float val  = xw[(size_t)s * 64 + cbase + j] * norm;
    atomic_add_f32(&h[(size_t)d * 32 + j], val);
}

// ---------------------------------------------------------------------------
// out[i] = relu(h[i]+b_near+b_sim) . W_lin + b_lin   (wave32 reduction)
// ---------------------------------------------------------------------------
__global__ void final_kernel(const float* __restrict__ h,
                             const float* __restrict__ bn,
                             const float* __restrict__ bs,
                             const float* __restrict__ wl,
                             const float* __restrict__ bl,
                             float* __restrict__ out, int n)
{
    const int t = blockIdx.x * blockDim.x + threadIdx.x;
    const int i = t >> 5, j = t & 31;
    if (i >= n) return;
    float v = h[(size_t)i * 32 + j] + bn[j] + bs[j];
    v = fmaxf(v, 0.0f) * wl[j];
    #pragma unroll
    for (int off = 16; off > 0; off >>= 1)
        v += __shfl_xor(v, off, 32);
    if (j == 0) out[i] = v + bl[0];
}

// ---------------------------------------------------------------------------
extern "C" void kernel_launch(void* const* d_in, const int* in_sizes, int n_in,
                              void* d_out, int out_size, void* d_ws, size_t ws_size,
                              hipStream_t stream)
{
    const float* x  = (const float*)d_in[0];
    const int*   en = (const int*)  d_in[1];   // [2,E] row-major: src then dst
    const int*   es = (const int*)  d_in[2];
    const float* Wn = (const float*)d_in[3];
    const float* bn = (const float*)d_in[4];
    const float* Ws = (const float*)d_in[5];
    const float* bs = (const float*)d_in[6];
    const float* Wl = (const float*)d_in[7];
    const float* bl = (const float*)d_in[8];
    float* out = (float*)d_out;

    const int N = in_sizes[0] / 64;     // 100000
    const int E = in_sizes[1] / 2;      // 3200000

    // workspace layout (floats): xw[N*64] | h[N*32] | dn[N] | ds[N]
    float* ws  = (float*)d_ws;
    float* xw  = ws;
    float* h   = xw + (size_t)N * 64;
    float* dn  = h  + (size_t)N * 32;
    float* ds  = dn + N;

    const int TPB = 256;

    // degrees (incl. self loop), then d^{-1/2}
    init_deg_kernel<<<(N + TPB - 1) / TPB, TPB, 0, stream>>>(dn, ds, N);
    count_deg_kernel<<<dim3((E + TPB - 1) / TPB, 2), TPB, 0, stream>>>(
        en + E, es + E, dn, ds, E);
    rsqrt_kernel<<<(N + TPB - 1) / TPB, TPB, 0, stream>>>(dn, ds, N);

    // fused xW GEMM via f32 WMMA: xw = x @ [W_near | W_similar]
    gemm_xw_kernel<<<(N + 15) / 16, 128, 0, stream>>>(x, Wn, Ws, xw, N);

    // self-loop init, then edge scatters (both conv outputs summed into h)
    init_h_kernel<<<(N * 32 + TPB - 1) / TPB, TPB, 0, stream>>>(xw, dn, ds, h, N);
    {
        const unsigned tot  = (unsigned)E * 32u;
        const unsigned grid = (tot + TPB - 1) / TPB;
        scatter_kernel<<<dim3(grid, 2), TPB, 0, stream>>>(en, es, xw, dn, ds, h, E);
    }

    // relu + linear head
    final_kernel<<<(N * 32 + TPB - 1) / TPB, TPB, 0, stream>>>(h, bn, bs, Wl, bl, out, N);
}